// DWTForward_42949672961261
// MI455X (gfx1250) — compile-verified
//
#include <hip/hip_runtime.h>
#include <hip/hip_bf16.h>

// Fused 3-level 2D Haar DWT (zero-pad mode; all sizes even -> pad == 0).
// Each 16x16 tile is transformed as Y = H^T * X * H via V_WMMA_F32_16X16X4_F32
// (8 WMMAs/tile, K accumulated in 4-chunks). Quadrants of Y are LL/HL/LH/HH.
//
// WMMA f32 layout assumptions (ISA 7.12.2, wave32):
//   A (16x4):  v0 = {K=0: lanes0-15, K=2: lanes16-31}, v1 = {K=1, K=3}; M = lane%16
//   B (4x16):  v0 = {K=0: lanes0-15, K=2: lanes16-31}, v1 = {K=1, K=3}; N = lane%16
//   C/D(16x16): VGPR r: lanes0-15 -> (M=r, N=lane), lanes16-31 -> (M=r+8, N=lane-16)
// Both "A = H^T" and "B = H" reduce to the per-lane value H[4k + 2*hs + v, lane%16].

typedef float v2f __attribute__((ext_vector_type(2)));
typedef float v4f __attribute__((ext_vector_type(4)));
typedef float v8f __attribute__((ext_vector_type(8)));

#define OFF_YH0 1048576u     // 8*32*64*64
#define OFF_YH1 51380224u    // + 8*32*3*256*256
#define OFF_YH2 63963136u    // + 8*32*3*128*128

// H[r][c]: analysis matrix column c<8 -> lo taps (f00,f01) at rows 2c,2c+1;
// c>=8 -> hi taps (f10,f11) at rows 2(c-8), 2(c-8)+1.
__device__ __forceinline__ float hval(int r, int c,
                                      float f00, float f01, float f10, float f11) {
  if (c < 8) {
    if (r == 2*c)     return f00;
    if (r == 2*c + 1) return f01;
    return 0.0f;
  }
  int c2 = c - 8;
  if (r == 2*c2)     return f10;
  if (r == 2*c2 + 1) return f11;
  return 0.0f;
}

// One 16x16 tile: returns Y = Hc^T * X * Hr in C/D layout.
// z: per-wave 16x17 LDS scratch for the intermediate Z = Hc^T * X.
__device__ __forceinline__ v8f haar_tile16(const float* src, int sstride,
                                           float* z,
                                           const v2f* Hc, const v2f* Hr,
                                           int hs, int lrow) {
  v8f zacc = {};
#pragma unroll
  for (int k = 0; k < 4; ++k) {          // Z = Hc^T * X   (height pass)
    const int r0 = 4*k + 2*hs;
    v2f b;
    b.x = src[(r0    ) * sstride + lrow]; // B chunk = rows of X
    b.y = src[(r0 + 1) * sstride + lrow];
    zacc = __builtin_amdgcn_wmma_f32_16x16x4_f32(false, Hc[k], false, b,
                                                 (short)0, zacc, false, false);
  }
#pragma unroll
  for (int r = 0; r < 8; ++r)            // spill Z (D layout -> row major LDS)
    z[(r + 8*hs)*17 + lrow] = zacc[r];

  v8f y = {};
#pragma unroll
  for (int k = 0; k < 4; ++k) {          // Y = Z * Hr     (width pass)
    const int c0 = 4*k + 2*hs;
    v2f a;
    a.x = z[lrow*17 + c0];               // A chunk = Z in A layout
    a.y = z[lrow*17 + c0 + 1];
    y = __builtin_amdgcn_wmma_f32_16x16x4_f32(false, a, false, Hr[k],
                                              (short)0, y, false, false);
  }
  return y;
}

// Scatter Y: LL quadrant -> llBlk (LDS); LH/HL/HH -> yhL global (band order lh,hl,hh).
__device__ __forceinline__ void emit_tile(const v8f& y, int hs, int lrow,
                                          float* llBlk, int llStride,
                                          float* yhL, int S,
                                          int img, int by, int bx) {
  if (hs == 0 && lrow < 8) {
#pragma unroll
    for (int r = 0; r < 8; ++r) llBlk[r*llStride + lrow] = y[r];
  } else {
    const int band = (hs == 0) ? 1 : ((lrow < 8) ? 0 : 2);  // HL=1, LH=0, HH=2
    const int c = lrow & 7;
    const size_t base = (((size_t)img*3 + band)*S + by)*(size_t)S + bx + c;
#pragma unroll
    for (int r = 0; r < 8; ++r)
      __builtin_nontemporal_store(y[r], &yhL[base + (size_t)r*S]);
  }
}

__global__ __launch_bounds__(128)
void haar_dwt3_kernel(const float* __restrict__ x,
                      const float* __restrict__ h0c, const float* __restrict__ h1c,
                      const float* __restrict__ h0r, const float* __restrict__ h1r,
                      float* __restrict__ out) {
  __shared__ __align__(16) float Xs[64][72];  // staged input, stride 72 (anti-conflict)
  __shared__ float Zs[4][16*17];              // per-wave Z scratch
  __shared__ float LL1[32][33];
  __shared__ float LL2[16][17];

  const int img  = blockIdx.y;        // n*32 + c
  const int tx64 = blockIdx.x & 7;
  const int ty64 = blockIdx.x >> 3;

  const int tid  = threadIdx.x;
  const int lane = tid & 31;
  const int wave = tid >> 5;
  const int hs   = lane >> 4;         // lane half
  const int lrow = lane & 15;

  // Filter taps (uniform scalar loads)
  const float c00 = h0c[0], c01 = h0c[1], c10 = h1c[0], c11 = h1c[1];
  const float r00 = h0r[0], r01 = h0r[1], r10 = h1r[0], r11 = h1r[1];
  v2f Hc[4], Hr[4];
#pragma unroll
  for (int k = 0; k < 4; ++k) {
    const int rr = 4*k + 2*hs;
    Hc[k].x = hval(rr,   lrow, c00, c01, c10, c11);
    Hc[k].y = hval(rr+1, lrow, c00, c01, c10, c11);
    Hr[k].x = hval(rr,   lrow, r00, r01, r10, r11);
    Hr[k].y = hval(rr+1, lrow, r00, r01, r10, r11);
  }

  // Stage 64x64 region -> LDS (coalesced float4, streaming)
  const float* src = x + ((size_t)img*512 + (size_t)ty64*64)*512 + (size_t)tx64*64;
  {
    const int c4 = (tid & 15) * 4;
    const int r0 = tid >> 4;
#pragma unroll
    for (int i = 0; i < 8; ++i) {
      const int r = r0 + i*8;
      const v4f v = __builtin_nontemporal_load((const v4f*)(src + (size_t)r*512 + c4));
      *(v4f*)&Xs[r][c4] = v;
    }
  }
  __syncthreads();

  float* zbuf = Zs[wave];
  float* yh0 = out + OFF_YH0;
  float* yh1 = out + OFF_YH1;
  float* yh2 = out + OFF_YH2;

  // ---- Level 1: 16 tiles of 16x16 (4 per wave) ----
  for (int t = wave; t < 16; t += 4) {
    const int tty = t >> 2, ttx = t & 3;
    v8f y = haar_tile16(&Xs[tty*16][ttx*16], 72, zbuf, Hc, Hr, hs, lrow);
    emit_tile(y, hs, lrow, &LL1[tty*8][ttx*8], 33, yh0, 256, img,
              ty64*32 + tty*8, tx64*32 + ttx*8);
  }
  __syncthreads();

  // ---- Level 2: 4 tiles of 16x16 from LL1 (1 per wave) ----
  {
    const int tty = wave >> 1, ttx = wave & 1;
    v8f y = haar_tile16(&LL1[tty*16][ttx*16], 33, zbuf, Hc, Hr, hs, lrow);
    emit_tile(y, hs, lrow, &LL2[tty*8][ttx*8], 17, yh1, 128, img,
              ty64*16 + tty*8, tx64*16 + ttx*8);
  }
  __syncthreads();

  // ---- Level 3: 1 tile from LL2 (wave 0 only; no barriers after) ----
  if (wave == 0) {
    v8f y = haar_tile16(&LL2[0][0], 17, zbuf, Hc, Hr, hs, lrow);
    if (hs == 0 && lrow < 8) {
      float* p = out + (size_t)img*64*64 + ((size_t)ty64*8)*64 + tx64*8 + lrow;
#pragma unroll
      for (int r = 0; r < 8; ++r)
        __builtin_nontemporal_store(y[r], &p[(size_t)r*64]);
    } else {
      const int band = (hs == 0) ? 1 : ((lrow < 8) ? 0 : 2);
      const int c = lrow & 7;
      const size_t base = (((size_t)img*3 + band)*64 + (size_t)ty64*8)*64
                        + (size_t)tx64*8 + c;
#pragma unroll
      for (int r = 0; r < 8; ++r)
        __builtin_nontemporal_store(y[r], &yh2[base + (size_t)r*64]);
    }
  }
}

extern "C" void kernel_launch(void* const* d_in, const int* in_sizes, int n_in,
                              void* d_out, int out_size, void* d_ws, size_t ws_size,
                              hipStream_t stream) {
  const float* x   = (const float*)d_in[0];
  const float* h0c = (const float*)d_in[1];
  const float* h1c = (const float*)d_in[2];
  const float* h0r = (const float*)d_in[3];
  const float* h1r = (const float*)d_in[4];
  float* out = (float*)d_out;

  // 8x8 regions of 64x64 per 512x512 image, 256 images
  dim3 grid(64, 256);
  haar_dwt3_kernel<<<grid, dim3(128), 0, stream>>>(x, h0c, h1c, h0r, h1r, out);
}